// ScatterMoeQwen3MoeSparseMoeBlock_91027536871548
// MI455X (gfx1250) — compile-verified
//
#include <hip/hip_runtime.h>
#include <math.h>

// MoE sparse block for MI455X (gfx1250, wave32).
// d_out = out[T*H] f32 ++ router_logits[T*E] f32.
// Required ws_size ~= 23.3 MB (compacted SwiGLU activation scratch).

typedef float v2f __attribute__((ext_vector_type(2)));
typedef float v8f __attribute__((ext_vector_type(8)));
typedef int   i32x4 __attribute__((vector_size(16)));

#define Tn   2048   // tokens
#define Hd   2048   // hidden
#define Id   1408   // intermediate
#define En   8      // experts
#define KC   32     // K chunk per LDS stage (Hd: 64 chunks, Id: 44 chunks)

// ---- workspace layout (bytes) ----
static constexpr size_t CNT_OFF    = 0;                        // E ints
static constexpr size_t HBASE_OFF  = 64;                       // E ints
static constexpr size_t TOKIDX_OFF = 128;                      // E*Tn ints
static constexpr size_t TOKW_OFF   = TOKIDX_OFF + (size_t)En * Tn * 4;
static constexpr size_t HS_OFF     = ((TOKW_OFF + (size_t)En * Tn * 4) + 255) & ~(size_t)255;
// HS: (2*Tn) rows x Id floats = 4096*1408*4 = 23,068,672 B

// ---------------------------------------------------------------------------
// CDNA5 async global->LDS copy (ASYNCcnt-tracked, bypasses VGPRs).
// Falls back to synchronous float4 staging if the builtin is unavailable.
// ---------------------------------------------------------------------------
#if __has_builtin(__builtin_amdgcn_global_load_async_to_lds_b128) && \
    __has_builtin(__builtin_amdgcn_s_wait_asynccnt)
__device__ __forceinline__ void cp16(void* lds, const void* g) {
  __builtin_amdgcn_global_load_async_to_lds_b128((i32x4*)g, (i32x4*)lds, 0, 0);
}
__device__ __forceinline__ void cp_wait() {
  __builtin_amdgcn_s_wait_asynccnt(0);
}
#else
__device__ __forceinline__ void cp16(void* lds, const void* g) {
  *(float4*)lds = *(const float4*)g;
}
__device__ __forceinline__ void cp_wait() {}
#endif

__device__ __forceinline__ v8f wmma4(v2f a, v2f b, v8f c) {
  return __builtin_amdgcn_wmma_f32_16x16x4_f32(false, a, false, b, (short)0, c,
                                               false, false);
}

// ---------------------------------------------------------------------------
// init: zero the out region and the per-expert counters
// ---------------------------------------------------------------------------
__global__ __launch_bounds__(256) void moe_init(float* __restrict__ out,
                                                int* __restrict__ cnt) {
  size_t idx = ((size_t)blockIdx.x * 256 + threadIdx.x) * 4;
  if (idx < (size_t)Tn * Hd) {
    float4 z = {0.f, 0.f, 0.f, 0.f};
    *(float4*)(out + idx) = z;
  }
  if (blockIdx.x == 0 && threadIdx.x < En) cnt[threadIdx.x] = 0;
}

// ---------------------------------------------------------------------------
// router: 1 block / token; 8 waves, one expert dot-product each (wave32)
// ---------------------------------------------------------------------------
__global__ __launch_bounds__(256) void moe_router(
    const float* __restrict__ x, const float* __restrict__ gw,
    float* __restrict__ logits, int* __restrict__ cnt,
    int* __restrict__ tokidx, float* __restrict__ tokw) {
  __shared__ float xs[Hd];
  __shared__ float lg[En];
  const int t = blockIdx.x;
  const int tid = threadIdx.x;
  for (int i = tid; i < Hd; i += 256) xs[i] = x[(size_t)t * Hd + i];
  __syncthreads();
  const int w = tid >> 5;      // expert handled by this wave
  const int lane = tid & 31;
  float s = 0.f;
  for (int k = lane; k < Hd; k += 32) s += xs[k] * gw[(size_t)w * Hd + k];
#pragma unroll
  for (int off = 16; off; off >>= 1) s += __shfl_xor(s, off, 32);
  if (lane == 0) lg[w] = s;
  __syncthreads();
  if (tid == 0) {
    int i1 = 0; float v1 = lg[0];
#pragma unroll
    for (int e = 1; e < En; ++e) if (lg[e] > v1) { v1 = lg[e]; i1 = e; }
    int i2 = (i1 == 0) ? 1 : 0; float v2 = lg[i2];
#pragma unroll
    for (int e = 0; e < En; ++e)
      if (e != i1 && lg[e] > v2) { v2 = lg[e]; i2 = e; }
    // normalized top-2 softmax weights; global Z cancels
    float w1 = 1.f / (1.f + __expf(v2 - v1));
    float w2 = 1.f - w1;
    int p1 = atomicAdd(&cnt[i1], 1);
    tokidx[i1 * Tn + p1] = t; tokw[i1 * Tn + p1] = w1;
    int p2 = atomicAdd(&cnt[i2], 1);
    tokidx[i2 * Tn + p2] = t; tokw[i2 * Tn + p2] = w2;
#pragma unroll
    for (int e = 0; e < En; ++e) logits[(size_t)t * En + e] = lg[e];
  }
}

// ---------------------------------------------------------------------------
// scan: exclusive prefix over 8 expert counts -> compacted H-scratch bases
// ---------------------------------------------------------------------------
__global__ void moe_scan(const int* __restrict__ cnt, int* __restrict__ hbase) {
  if (threadIdx.x == 0) {
    int s = 0;
    for (int e = 0; e < En; ++e) { hbase[e] = s; s += cnt[e]; }
  }
}

// ---------------------------------------------------------------------------
// GEMM1: per expert, gathered tokens. H = silu(X*Wg) * (X*Wu)
// block 256 thr = 8 waves; block tile 64 tok x 64 I-cols; each wave 32Mx16N
// double-buffered LDS filled by async global->LDS b128 copies.
// LDS pitches: A 36 dw (conflict-free frag reads), W 80 dw (80%64==16 ->
// the 4 K-rows of a fragment pair hit disjoint 16-bank groups).
// ---------------------------------------------------------------------------
__global__ __launch_bounds__(256) void moe_gemm1(
    const float* __restrict__ x, const float* __restrict__ wg,
    const float* __restrict__ wu, const int* __restrict__ cnt,
    const int* __restrict__ hbase, const int* __restrict__ tokidx,
    float* __restrict__ Hs) {
  const int e = blockIdx.z;
  const int mc = cnt[e];
  const int m0 = blockIdx.y * 64;
  if (m0 >= mc) return;
  const int n0 = blockIdx.x * 64;

  __shared__ float Xs[2][64][36];   // [buf][m][k]
  __shared__ float Gs[2][KC][80];   // [buf][k][n]
  __shared__ float Us[2][KC][80];

  const int tid = threadIdx.x;
  const int wid = tid >> 5;
  const int lane = tid & 31;
  const int half = lane >> 4;
  const int lr = lane & 15;
  const int wm = wid >> 2;   // 0..1 -> 32-row band
  const int wn = wid & 3;    // 0..3 -> 16-col band

  // gathered token row per X-transfer this thread performs (2 per chunk)
  int tokr[2];
#pragma unroll
  for (int p = 0; p < 2; ++p) {
    int r = (tid + 256 * p) >> 3;              // 64 rows, 8 x16B per row
    int row = m0 + r; if (row > mc - 1) row = mc - 1;
    tokr[p] = tokidx[e * Tn + row];
  }

  auto issue = [&](int buf, int k0) {
#pragma unroll
    for (int p = 0; p < 2; ++p) {              // X tile: 64 x KC
      int j = tid + 256 * p;
      int r = j >> 3, c4 = (j & 7) << 2;
      cp16(&Xs[buf][r][c4], x + (size_t)tokr[p] * Hd + k0 + c4);
    }
#pragma unroll
    for (int p = 0; p < 2; ++p) {              // W tiles: KC x 64 (row-major)
      int j = tid + 256 * p;
      int r = j >> 4, c4 = (j & 15) << 2;
      size_t goff = ((size_t)e * Hd + k0 + r) * Id + n0 + c4;
      cp16(&Gs[buf][r][c4], wg + goff);
      cp16(&Us[buf][r][c4], wu + goff);
    }
  };

  v8f accG[2], accU[2];
#pragma unroll
  for (int s2 = 0; s2 < 2; ++s2) {
    accG[s2] = {0.f, 0.f, 0.f, 0.f, 0.f, 0.f, 0.f, 0.f};
    accU[s2] = {0.f, 0.f, 0.f, 0.f, 0.f, 0.f, 0.f, 0.f};
  }

  issue(0, 0);
  cp_wait();
  __syncthreads();
  int buf = 0;
  for (int k0 = 0; k0 < Hd; k0 += KC) {
    if (k0 + KC < Hd) issue(buf ^ 1, k0 + KC);   // prefetch next chunk (async)
#pragma unroll
    for (int kk = 0; kk < KC; kk += 4) {
      const int kr = kk + 2 * half;
      // A frags: lanes<16 hold {K0,K1}, lanes>=16 {K2,K3}
      v2f a0 = *(const v2f*)&Xs[buf][wm * 32 + lr][kr];
      v2f a1 = *(const v2f*)&Xs[buf][wm * 32 + 16 + lr][kr];
      v2f bg = {Gs[buf][kr][wn * 16 + lr], Gs[buf][kr + 1][wn * 16 + lr]};
      v2f bu = {Us[buf][kr][wn * 16 + lr], Us[buf][kr + 1][wn * 16 + lr]};
      accG[0] = wmma4(a0, bg, accG[0]);
      accG[1] = wmma4(a1, bg, accG[1]);
      accU[0] = wmma4(a0, bu, accU[0]);
      accU[1] = wmma4(a1, bu, accU[1]);
    }
    cp_wait();          // next chunk's async writes landed
    __syncthreads();    // all waves done (reads of buf, writes of buf^1)
    buf ^= 1;
  }

  const int hb = hbase[e];
#pragma unroll
  for (int s2 = 0; s2 < 2; ++s2) {
#pragma unroll
    for (int i = 0; i < 8; ++i) {   // C/D layout: M = i + 8*half, N = lr
      int ml = wm * 32 + s2 * 16 + i + 8 * half;
      int nl = wn * 16 + lr;
      if (m0 + ml < mc) {
        float g = accG[s2][i], u = accU[s2][i];
        float h = (g / (1.f + __expf(-g))) * u;   // SiLU(g) * u
        Hs[(size_t)(hb + m0 + ml) * Id + n0 + nl] = h;
      }
    }
  }
}

// ---------------------------------------------------------------------------
// GEMM2: Y = H * Wd, scaled by routing weight, atomically combined into out
// ---------------------------------------------------------------------------
__global__ __launch_bounds__(256) void moe_gemm2(
    const float* __restrict__ Hs, const float* __restrict__ wd,
    const int* __restrict__ cnt, const int* __restrict__ hbase,
    const int* __restrict__ tokidx, const float* __restrict__ tokw,
    float* __restrict__ out) {
  const int e = blockIdx.z;
  const int mc = cnt[e];
  const int m0 = blockIdx.y * 64;
  if (m0 >= mc) return;
  const int n0 = blockIdx.x * 64;   // over Hd

  __shared__ float As[2][64][36];
  __shared__ float Ws[2][KC][80];

  const int tid = threadIdx.x;
  const int wid = tid >> 5;
  const int lane = tid & 31;
  const int half = lane >> 4;
  const int lr = lane & 15;
  const int wm = wid >> 2;
  const int wn = wid & 3;
  const int hb = hbase[e];

  auto issue = [&](int buf, int k0) {
#pragma unroll
    for (int p = 0; p < 2; ++p) {              // H tile rows are contiguous
      int j = tid + 256 * p;
      int r = j >> 3, c4 = (j & 7) << 2;
      int row = m0 + r; if (row > mc - 1) row = mc - 1;
      cp16(&As[buf][r][c4], Hs + (size_t)(hb + row) * Id + k0 + c4);
    }
#pragma unroll
    for (int p = 0; p < 2; ++p) {
      int j = tid + 256 * p;
      int r = j >> 4, c4 = (j & 15) << 2;
      cp16(&Ws[buf][r][c4], wd + ((size_t)e * Id + k0 + r) * Hd + n0 + c4);
    }
  };

  v8f acc[2];
  acc[0] = {0.f, 0.f, 0.f, 0.f, 0.f, 0.f, 0.f, 0.f};
  acc[1] = {0.f, 0.f, 0.f, 0.f, 0.f, 0.f, 0.f, 0.f};

  issue(0, 0);
  cp_wait();
  __syncthreads();
  int buf = 0;
  for (int k0 = 0; k0 < Id; k0 += KC) {   // 1408 = 44 * 32
    if (k0 + KC < Id) issue(buf ^ 1, k0 + KC);
#pragma unroll
    for (int kk = 0; kk < KC; kk += 4) {
      const int kr = kk + 2 * half;
      v2f a0 = *(const v2f*)&As[buf][wm * 32 + lr][kr];
      v2f a1 = *(const v2f*)&As[buf][wm * 32 + 16 + lr][kr];
      v2f b  = {Ws[buf][kr][wn * 16 + lr], Ws[buf][kr + 1][wn * 16 + lr]};
      acc[0] = wmma4(a0, b, acc[0]);
      acc[1] = wmma4(a1, b, acc[1]);
    }
    cp_wait();
    __syncthreads();
    buf ^= 1;
  }

#pragma unroll
  for (int s2 = 0; s2 < 2; ++s2) {
#pragma unroll
    for (int i = 0; i < 8; ++i) {
      int ml = wm * 32 + s2 * 16 + i + 8 * half;
      int nl = wn * 16 + lr;
      int row = m0 + ml;
      if (row < mc) {
        int tok = tokidx[e * Tn + row];
        float wgt = tokw[e * Tn + row];
        // exactly 2 commutative f32 adds per out element -> deterministic
        atomicAdd(&out[(size_t)tok * Hd + n0 + nl], wgt * acc[s2][i]);
      }
    }
  }
}

// ---------------------------------------------------------------------------
extern "C" void kernel_launch(void* const* d_in, const int* in_sizes, int n_in,
                              void* d_out, int out_size, void* d_ws,
                              size_t ws_size, hipStream_t stream) {
  const float* x  = (const float*)d_in[0];   // [1,2048,2048]
  const float* gw = (const float*)d_in[1];   // [8,2048]
  const float* wg = (const float*)d_in[2];   // [8,2048,1408]
  const float* wu = (const float*)d_in[3];   // [8,2048,1408]
  const float* wd = (const float*)d_in[4];   // [8,1408,2048]
  float* out    = (float*)d_out;
  float* logits = out + (size_t)Tn * Hd;

  char* ws = (char*)d_ws;
  int*   cnt    = (int*)(ws + CNT_OFF);
  int*   hbase  = (int*)(ws + HBASE_OFF);
  int*   tokidx = (int*)(ws + TOKIDX_OFF);
  float* tokw   = (float*)(ws + TOKW_OFF);
  float* Hs     = (float*)(ws + HS_OFF);

  moe_init<<<(Tn * Hd / 4 + 255) / 256, 256, 0, stream>>>(out, cnt);
  moe_router<<<Tn, 256, 0, stream>>>(x, gw, logits, cnt, tokidx, tokw);
  moe_scan<<<1, 32, 0, stream>>>(cnt, hbase);
  dim3 g1(Id / 64, Tn / 64, En);   // (22, 32, 8)
  moe_gemm1<<<g1, 256, 0, stream>>>(x, wg, wu, cnt, hbase, tokidx, Hs);
  dim3 g2(Hd / 64, Tn / 64, En);   // (32, 32, 8)
  moe_gemm2<<<g2, 256, 0, stream>>>(Hs, wd, cnt, hbase, tokidx, tokw, out);
}